// MixedOp_35098472743519
// MI455X (gfx1250) — compile-verified
//
#include <hip/hip_runtime.h>
#include <cstdint>

#define B_DIM 32
#define L_DIM 1024
#define O_DIM 4
#define D_DIM 256
#define ROW_ELEMS (O_DIM * D_DIM)   // 1024 channels per output row
#define LP2 (L_DIM + 2)             // 1026 output rows per batch

// One block per output row (b, pos). 256 threads x float4 = 1024 floats.
__global__ __launch_bounds__(256)
void scale_pad_rows_kernel(const float* __restrict__ x,
                           const float* __restrict__ w,
                           const long long* __restrict__ lengths,
                           float* __restrict__ out)
{
    __shared__ __align__(16) float tile[ROW_ELEMS];   // 4 KB stage buffer

    const int t   = threadIdx.x;
    const int row = blockIdx.x;           // row = b * 1026 + pos
    const int b   = row / LP2;
    const int pos = row - b * LP2;

    float4* orow = reinterpret_cast<float4*>(out + (size_t)row * ROW_ELEMS) + t;

    const long long len = lengths[b];

    if (pos == 0) {                        // CLS row
        *orow = make_float4(1.f, 1.f, 1.f, 1.f);
        return;
    }
    if ((long long)pos == len + 1) {       // SEP row
        *orow = make_float4(2.f, 2.f, 2.f, 2.f);
        return;
    }
    if ((long long)pos > len) {            // zero tail — no load needed
        *orow = make_float4(0.f, 0.f, 0.f, 0.f);
        return;
    }

    // --- token row: async-stage the contiguous 4 KB x-row into LDS ---
    const float*   xrow   = x + ((size_t)b * L_DIM + (size_t)(pos - 1)) * ROW_ELEMS;
    const uint64_t gaddr  = (uint64_t)(uintptr_t)(xrow + t * 4);
    const uint32_t ldsoff = (uint32_t)(uintptr_t)(&tile[0]) + (uint32_t)t * 16u;

    // gfx1250 async tensor/LDS copy path; tracked by ASYNCcnt.
    asm volatile("global_load_async_to_lds_b128 %0, %1, off"
                 :: "v"(ldsoff), "v"(gaddr)
                 : "memory");

    // Softmax over the 4 gate weights (redundant per thread; trivially cheap
    // next to the 32 B/thread of HBM traffic). Overlaps the async load.
    const float w0 = w[0], w1 = w[1], w2 = w[2], w3 = w[3];
    const float m  = fmaxf(fmaxf(w0, w1), fmaxf(w2, w3));
    const float e0 = __expf(w0 - m);
    const float e1 = __expf(w1 - m);
    const float e2 = __expf(w2 - m);
    const float e3 = __expf(w3 - m);
    const float inv = 1.f / (e0 + e1 + e2 + e3);
    const int   o   = t >> 6;              // (t*4)/256 : which of the 4 gates
    const float s   = ((o == 0) ? e0 : (o == 1) ? e1 : (o == 2) ? e2 : e3) * inv;

    // Wait for this wave's async copy to land in LDS. Per-wave counter and
    // each lane reads only its own slot, so no workgroup barrier is needed.
    asm volatile("s_wait_asynccnt 0" ::: "memory");

    const float4 v = *reinterpret_cast<const float4*>(&tile[t * 4]);
    *orow = make_float4(v.x * s, v.y * s, v.z * s, v.w * s);
}

extern "C" void kernel_launch(void* const* d_in, const int* in_sizes, int n_in,
                              void* d_out, int out_size, void* d_ws, size_t ws_size,
                              hipStream_t stream) {
    const float*     x       = (const float*)d_in[0];
    const float*     w       = (const float*)d_in[1];
    const long long* lengths = (const long long*)d_in[2];
    float*           out     = (float*)d_out;

    dim3 grid(B_DIM * LP2);   // 32832 row-blocks
    dim3 block(256);
    hipLaunchKernelGGL(scale_pad_rows_kernel, grid, block, 0, stream,
                       x, w, lengths, out);
}